// recurrent_unrolling_867583394079
// MI455X (gfx1250) — compile-verified
//
#include <hip/hip_runtime.h>
#include <cstdint>
#include <cstddef>

// Problem constants (fixed by the reference file)
#define M_NODES  1000
#define L_EDGES  499500
#define BATCH    16
#define T_UNROLL 15

#define NSEG   8
#define SEGLEN (M_NODES / NSEG)   // 125
#define MPAD   1024
#define CH     4096               // edges per block in the edge pass

// ---------------------------------------------------------------------------
// CDNA5 async global->LDS staging (guarded; falls back to plain loads)
// ---------------------------------------------------------------------------
#if __has_builtin(__builtin_amdgcn_global_load_async_to_lds_b128) && \
    __has_builtin(__builtin_amdgcn_s_wait_asynccnt)
#define ATHENA_ASYNC_LDS 1
#else
#warning "CDNA5 async global->LDS builtins unavailable; using plain LDS fill"
#endif

typedef int v4i __attribute__((vector_size(16)));
typedef __attribute__((address_space(1))) v4i* gv4i_p;  // ptr to v4i in global AS
typedef __attribute__((address_space(3))) v4i* lv4i_p;  // ptr to v4i in LDS AS

__device__ __forceinline__ void lds_fill_row(float* lds, const float* src,
                                             int n, int tid, int nthr) {
#ifdef ATHENA_ASYNC_LDS
    const int nvec = n >> 2;  // n is a multiple of 4 (1000)
    for (int k = tid; k < nvec; k += nthr) {
        __builtin_amdgcn_global_load_async_to_lds_b128(
            (gv4i_p)(src + 4 * k),
            (lv4i_p)(lds + 4 * k),
            0, 0);
    }
#else
    for (int k = tid; k < n; k += nthr) lds[k] = src[k];
#endif
}

__device__ __forceinline__ void lds_fill_wait() {
#ifdef ATHENA_ASYNC_LDS
    __builtin_amdgcn_s_wait_asynccnt(0);
#endif
    __syncthreads();
}

// ---------------------------------------------------------------------------
// Edge indexing helpers: edge e of triu(m,k=1) connects (i, j), i<j.
// off(i) = i*(m-1) - i*(i-1)/2 ; disc = (2m-1)^2 - 8e equals (2m-1-2i)^2
// exactly at each row start, so a f32 sqrt + integer fixup recovers i exactly.
// ---------------------------------------------------------------------------
__device__ __forceinline__ int rowoff(int i) {
    return i * (M_NODES - 1) - ((i * (i - 1)) >> 1);
}

__device__ __forceinline__ void edge_ij(int e, int& io, int& jo) {
    const int disc = 3996001 - 8 * e;            // (2*1000-1)^2 - 8e, > 0
    const float s  = sqrtf((float)disc);
    int i = (int)((1999.0f - s) * 0.5f);
    i = i < 0 ? 0 : (i > M_NODES - 2 ? M_NODES - 2 : i);
    while (rowoff(i) > e)       --i;             // at most 1 step
    while (rowoff(i + 1) <= e)  ++i;             // at most 1 step
    io = i;
    jo = e - rowoff(i) + i + 1;
}

// ---------------------------------------------------------------------------
// Zero the node state v (scratch is poisoned with 0xAA by the harness)
// ---------------------------------------------------------------------------
__global__ __launch_bounds__(256) void k_zero_v(float* v) {
    const int idx = blockIdx.x * 256 + threadIdx.x;
    if (idx < BATCH * M_NODES) v[idx] = 0.0f;
}

// ---------------------------------------------------------------------------
// Node pass: for each (batch b, node n, segment of "other" nodes) accumulate
//   Sw_part = sum over incident edges of w_e
//   Sp_part = sum over incident edges of p1_e
// Split into branch-free lower (other<n, column side, incremental stride) and
// upper (other>n, contiguous row side) loops. Every node has exactly m-1
// incident edges -> perfectly balanced. Fixed-order partial sums, no atomics
// -> bitwise deterministic. w for iteration `it` lives in output slice it-1.
// ---------------------------------------------------------------------------
template<bool ZW>
__global__ __launch_bounds__(256) void k_node_pass(
    const float* __restrict__ z,
    const float* __restrict__ out,        // w source: slice it-1
    const float* __restrict__ vcur,
    float* __restrict__ Pw, float* __restrict__ Pp,
    const float* __restrict__ gn_p, const float* __restrict__ beta_p,
    int it)
{
    __shared__ float v_l[M_NODES];

    const int tid = threadIdx.x;
    const int nb  = blockIdx.x;   // node block (0..3)
    const int seg = blockIdx.y;   // 0..NSEG-1
    const int b   = blockIdx.z;   // batch

    const float gn   = gn_p[0];
    const float beta = beta_p[0];
    const float c1   = 1.0f - 2.0f * gn * beta;
    const float tg   = 2.0f * gn;

    lds_fill_row(v_l, vcur + b * M_NODES, M_NODES, tid, 256);
    lds_fill_wait();

    const int n = nb * 256 + tid;

    if (n < M_NODES) {
        const float* zrow = z + (size_t)b * L_EDGES;
        const float* wrow = out + (size_t)b * ((size_t)T_UNROLL * L_EDGES)
                                + (size_t)(ZW ? 0 : (it - 1)) * L_EDGES;
        const float vn   = v_l[n];
        const int   offn = rowoff(n);
        const int   o0   = seg * SEGLEN;
        const int   o1   = o0 + SEGLEN;

        float accw = 0.0f, accp = 0.0f;

        // ---- lower triangle side: other in [o0, min(n, o1)) ----
        {
            const int lo_end = (n < o1) ? ((n > o0) ? n : o0) : o1;
            int e  = rowoff(o0) + (n - o0 - 1);
            int de = (M_NODES - 2) - o0;          // e(o+1)-e(o) = 998 - o
            for (int o = o0; o < lo_end; ++o) {
                const float wv = ZW ? 0.0f : wrow[e];
                const float zv = zrow[e];
                const float y1 = wv * c1 - gn * (vn + v_l[o]);
                accp += fmaxf(0.0f, y1 - tg * zv);
                accw += wv;
                e += de; --de;
            }
        }
        // ---- upper triangle side: other in [max(n+1, o0), o1), contiguous e
        {
            const int hi_beg = (n + 1 > o0) ? (n + 1) : o0;
            int e = offn + (hi_beg - n - 1);
            for (int o = hi_beg; o < o1; ++o, ++e) {
                const float wv = ZW ? 0.0f : wrow[e];
                const float zv = zrow[e];
                const float y1 = wv * c1 - gn * (vn + v_l[o]);
                accp += fmaxf(0.0f, y1 - tg * zv);
                accw += wv;
            }
        }

        const int pidx = (seg * BATCH + b) * MPAD + n;
        Pw[pidx] = accw;
        Pp[pidx] = accp;
    }
}

// ---------------------------------------------------------------------------
// Node update (tiny): reduce segment partials in fixed order, then
//   y2 = v + gn*Sw ; p2 = (y2 - sqrt(max(y2^2 + 4 gn alpha, 1e-8)))/2
//   v_next = v - y2 + p2 + gn*Sp
// ---------------------------------------------------------------------------
__global__ __launch_bounds__(256) void k_node_update(
    const float* __restrict__ vcur,
    const float* __restrict__ Pw, const float* __restrict__ Pp,
    float* __restrict__ vnxt, float* __restrict__ p2o,
    const float* __restrict__ gn_p, const float* __restrict__ alpha_p)
{
    const int n = blockIdx.x * 256 + threadIdx.x;
    const int b = blockIdx.y;
    if (n >= M_NODES) return;

    const float gn    = gn_p[0];
    const float alpha = alpha_p[0];

    float Sw = 0.0f, Sp = 0.0f;
#pragma unroll
    for (int s = 0; s < NSEG; ++s) {
        const int pidx = (s * BATCH + b) * MPAD + n;
        Sw += Pw[pidx];
        Sp += Pp[pidx];
    }

    const int idx = b * M_NODES + n;
    const float v  = vcur[idx];
    const float y2 = v + gn * Sw;
    const float up = fmaxf(y2 * y2 + 4.0f * gn * alpha, 1e-8f);
    const float p2 = (y2 - sqrtf(up)) * 0.5f;
    vnxt[idx] = v - y2 + p2 + gn * Sp;
    p2o[idx]  = p2;
}

// ---------------------------------------------------------------------------
// Edge pass: per edge recompute y1/p1, apply q1 with LDS-resident p2, write
// new w directly into the output slice `it` (which IS the w state).
// ---------------------------------------------------------------------------
template<bool ZW>
__global__ __launch_bounds__(256) void k_edge_pass(
    const float* __restrict__ z,
    float* __restrict__ out,
    const float* __restrict__ vcur,
    const float* __restrict__ p2,
    const float* __restrict__ gn_p, const float* __restrict__ beta_p,
    int it)
{
    __shared__ float v_l[M_NODES];
    __shared__ float p_l[M_NODES];

    const int tid = threadIdx.x;
    const int b   = blockIdx.y;

    const float gn   = gn_p[0];
    const float beta = beta_p[0];
    const float c1   = 1.0f - 2.0f * gn * beta;
    const float tg   = 2.0f * gn;

    lds_fill_row(v_l, vcur + b * M_NODES, M_NODES, tid, 256);
    lds_fill_row(p_l, p2   + b * M_NODES, M_NODES, tid, 256);
    lds_fill_wait();

    const size_t bout = (size_t)b * ((size_t)T_UNROLL * L_EDGES);
    const float* zrow = z + (size_t)b * L_EDGES;
    const float* wrow = out + bout + (size_t)(ZW ? 0 : (it - 1)) * L_EDGES;
    float*       orow = out + bout + (size_t)it * L_EDGES;

    const int base = blockIdx.x * CH;
#pragma unroll
    for (int k = 0; k < CH / 256; ++k) {
        const int e = base + tid + k * 256;
        if (e >= L_EDGES) break;
        int i, j;
        edge_ij(e, i, j);
        const float wv  = ZW ? 0.0f : wrow[e];
        const float zv  = zrow[e];
        const float y1  = wv * c1 - gn * (v_l[i] + v_l[j]);
        const float p1v = fmaxf(0.0f, y1 - tg * zv);
        const float q1  = p1v * c1 - gn * (p_l[i] + p_l[j]);
        orow[e] = wv - y1 + q1;
    }
}

// ---------------------------------------------------------------------------
// Host-side launch: 1 init + 3 kernels per unroll step, all on `stream`.
// Scratch layout (floats): v0[16000] v1[16000] p2[16000] Pw[8*16*1024]
// Pp[8*16*1024]  => ~1.24 MB total.
// ---------------------------------------------------------------------------
extern "C" void kernel_launch(void* const* d_in, const int* in_sizes, int n_in,
                              void* d_out, int out_size, void* d_ws, size_t ws_size,
                              hipStream_t stream)
{
    (void)in_sizes; (void)n_in; (void)out_size; (void)ws_size;

    const float* z     = (const float*)d_in[0];
    const float* gn    = (const float*)d_in[1];
    const float* alpha = (const float*)d_in[2];
    const float* beta  = (const float*)d_in[3];
    // d_in[4] = num_unroll (device int) -- fixed at 15 by the reference.
    float* out = (float*)d_out;

    float* ws = (float*)d_ws;
    float* v0 = ws;
    float* v1 = v0 + BATCH * M_NODES;
    float* p2 = v1 + BATCH * M_NODES;
    float* Pw = p2 + BATCH * M_NODES;
    float* Pp = Pw + NSEG * BATCH * MPAD;

    k_zero_v<<<dim3((BATCH * M_NODES + 255) / 256), 256, 0, stream>>>(v0);

    float* vcur = v0;
    float* vnxt = v1;
    const int nchunks = (L_EDGES + CH - 1) / CH;   // 122
    const int nodeblk = (M_NODES + 255) / 256;     // 4

    for (int it = 0; it < T_UNROLL; ++it) {
        if (it == 0) {
            k_node_pass<true><<<dim3(nodeblk, NSEG, BATCH), 256, 0, stream>>>(
                z, out, vcur, Pw, Pp, gn, beta, it);
        } else {
            k_node_pass<false><<<dim3(nodeblk, NSEG, BATCH), 256, 0, stream>>>(
                z, out, vcur, Pw, Pp, gn, beta, it);
        }

        k_node_update<<<dim3(nodeblk, BATCH), 256, 0, stream>>>(
            vcur, Pw, Pp, vnxt, p2, gn, alpha);

        if (it == 0) {
            k_edge_pass<true><<<dim3(nchunks, BATCH), 256, 0, stream>>>(
                z, out, vcur, p2, gn, beta, it);
        } else {
            k_edge_pass<false><<<dim3(nchunks, BATCH), 256, 0, stream>>>(
                z, out, vcur, p2, gn, beta, it);
        }

        float* t = vcur; vcur = vnxt; vnxt = t;
    }
}